// GAT_44246753083466
// MI455X (gfx1250) — compile-verified
//
#include <hip/hip_runtime.h>

typedef float v2f __attribute__((ext_vector_type(2)));
typedef float v8f __attribute__((ext_vector_type(8)));

#define NEG_SLOPE 0.2f
#define EPS_DENOM 1e-16f

// ---------------------------------------------------------------------------
// GEMM: Y[N x NCOL] = X[N x 128] * W[128 x NCOL] using V_WMMA_F32_16X16X4_F32.
// One block = 16 rows of Y; one wave (32 lanes) = one 16x16 output tile.
// A-frag (16x4 f32, 2 VGPR): lanes 0-15 -> M=lane, K=k0+{0,1}; lanes 16-31 -> K=k0+{2,3}
// B-frag (4x16 f32, 2 VGPR): lanes 0-15 -> N=lane, K=k0+{0,1}; lanes 16-31 -> K=k0+{2,3}
// D (16x16 f32, 8 VGPR): VGPR r -> row r + (lane>=16 ? 8 : 0), col = lane&15
// ---------------------------------------------------------------------------
template <int NCOL>
__global__ __launch_bounds__(NCOL * 2) void gemm_wmma_f32(
    const float* __restrict__ X, const float* __restrict__ W,
    float* __restrict__ Y, int N) {
  constexpr int K = 128;
  constexpr int KV = K / 4;              // row length in float4
  __shared__ float As[16][K + 4];

  const int rowbase = blockIdx.x * 16;
  const bool full_tile = (rowbase + 16 <= N);

  // stage X tile into LDS as float4 (global_load_b128 / ds_store_b128)
  const float4* Xv = (const float4*)X;
  for (int i = threadIdx.x; i < 16 * KV; i += blockDim.x) {
    int r = i >> 5;                      // i / KV   (KV == 32)
    int c4 = i & (KV - 1);               // i % KV
    int gr = rowbase + r;
    float4 v = (full_tile || gr < N) ? Xv[(long long)gr * KV + c4]
                                     : make_float4(0.f, 0.f, 0.f, 0.f);
    *(float4*)&As[r][c4 * 4] = v;
  }
  __syncthreads();

  const int wave = threadIdx.x >> 5;
  const int lane = threadIdx.x & 31;
  const int colbase = wave * 16;
  const int m = lane & 15;               // A row / B col / D col
  const int kb = (lane >> 4) * 2;        // K sub-offset: 0 or 2

  v8f acc = {};
#pragma unroll 4
  for (int k0 = 0; k0 < K; k0 += 4) {
    v2f a, b;
    a.x = As[m][k0 + kb];
    a.y = As[m][k0 + kb + 1];
    b.x = W[(k0 + kb) * NCOL + colbase + m];
    b.y = W[(k0 + kb + 1) * NCOL + colbase + m];
    acc = __builtin_amdgcn_wmma_f32_16x16x4_f32(false, a, false, b,
                                                (short)0, acc, false, false);
  }

  const int rhalf = (lane >> 4) * 8;
  if (full_tile) {
#pragma unroll
    for (int r = 0; r < 8; ++r)
      Y[(long long)(rowbase + r + rhalf) * NCOL + colbase + m] = acc[r];
  } else {
#pragma unroll
    for (int r = 0; r < 8; ++r) {
      int gr = rowbase + r + rhalf;
      if (gr < N) Y[(long long)gr * NCOL + colbase + m] = acc[r];
    }
  }
}

// ---------------------------------------------------------------------------
// Per-node attention scores: as[n,h] = <h[n,h,:], att_src[h,:]>, ad likewise.
// H, C compile-time powers of two.
// ---------------------------------------------------------------------------
template <int H, int C>
__global__ void attn_scores_kernel(const float* __restrict__ h,
                                   const float* __restrict__ att_s,
                                   const float* __restrict__ att_d,
                                   float* __restrict__ as_,
                                   float* __restrict__ ad_,
                                   int N) {
  int idx = blockIdx.x * blockDim.x + threadIdx.x;
  if (idx >= N * H) return;
  int n = idx / H, hh = idx & (H - 1);
  const float* hp = h + (long long)n * (H * C) + hh * C;
  float s = 0.f, d = 0.f;
#pragma unroll 8
  for (int c = 0; c < C; ++c) {
    float v = hp[c];
    s += v * att_s[hh * C + c];
    d += v * att_d[hh * C + c];
  }
  as_[idx] = s;
  ad_[idx] = d;
}

// Order-preserving float<->uint key for atomic max (key 0 == -inf sentinel).
__device__ __forceinline__ unsigned f2key(float f) {
  unsigned b = __float_as_uint(f);
  return (b & 0x80000000u) ? ~b : (b | 0x80000000u);
}
__device__ __forceinline__ float key2f(unsigned u) {
  unsigned b = (u & 0x80000000u) ? (u & 0x7fffffffu) : ~u;
  return __uint_as_float(b);
}

// Pass A: alpha_raw = leakyrelu(as[src]+ad[dst]); segment max into mkey[dst].
template <int H>
__global__ void edge_alpha_max_kernel(const int* __restrict__ src,
                                      const int* __restrict__ dst,
                                      const float* __restrict__ as_,
                                      const float* __restrict__ ad_,
                                      float* __restrict__ alpha,
                                      unsigned* __restrict__ mkey,
                                      int E, int Etot) {
  unsigned idx = blockIdx.x * blockDim.x + threadIdx.x;   // < Etot*H <= 6.8M
  if (idx >= (unsigned)Etot * H) return;
  int e = idx / H, hh = idx & (H - 1);
  int s = (e < E) ? src[e] : (e - E);
  int d = (e < E) ? dst[e] : (e - E);
  float a = as_[s * H + hh] + ad_[d * H + hh];
  a = (a > 0.f) ? a : NEG_SLOPE * a;
  alpha[idx] = a;
  atomicMax(&mkey[d * H + hh], f2key(a));
}

// Pass B: alpha = exp(alpha - m[dst]); segment sum into denom[dst].
template <int H>
__global__ void edge_exp_sum_kernel(const int* __restrict__ dst,
                                    const unsigned* __restrict__ mkey,
                                    float* __restrict__ alpha,
                                    float* __restrict__ denom,
                                    int E, int Etot) {
  unsigned idx = blockIdx.x * blockDim.x + threadIdx.x;
  if (idx >= (unsigned)Etot * H) return;
  int e = idx / H, hh = idx & (H - 1);
  int d = (e < E) ? dst[e] : (e - E);
  float a = __expf(alpha[idx] - key2f(mkey[d * H + hh]));
  alpha[idx] = a;
  atomicAdd(&denom[d * H + hh], a);
}

// Pass B2: normalize alpha by denom[dst] + eps.
template <int H>
__global__ void edge_norm_kernel(const int* __restrict__ dst,
                                 const float* __restrict__ denom,
                                 float* __restrict__ alpha,
                                 int E, int Etot) {
  unsigned idx = blockIdx.x * blockDim.x + threadIdx.x;
  if (idx >= (unsigned)Etot * H) return;
  int e = idx / H, hh = idx & (H - 1);
  int d = (e < E) ? dst[e] : (e - E);
  alpha[idx] = alpha[idx] / (denom[d * H + hh] + EPS_DENOM);
}

// Pass C: out[dst, ch] += alpha[e, ch/C] * h[src, ch].  F = H*C threads/edge.
template <int H, int C>
__global__ void edge_aggregate_kernel(const int* __restrict__ src,
                                      const int* __restrict__ dst,
                                      const float* __restrict__ alpha,
                                      const float* __restrict__ h,
                                      float* __restrict__ out,
                                      int E, int Etot) {
  constexpr int F = H * C;
  long long gid = blockIdx.x * (long long)blockDim.x + threadIdx.x;
  if (gid >= (long long)Etot * F) return;
  int e = (int)(gid >> (31 - __builtin_clz(F)));          // gid / F (F pow2)
  int ch = (int)gid & (F - 1);
  int s = (e < E) ? src[e] : (e - E);
  int d = (e < E) ? dst[e] : (e - E);
  int hh = ch / C;                                        // pow2 -> shift
  float a = alpha[(long long)e * H + hh];
  atomicAdd(&out[(long long)d * F + ch], a * h[(long long)s * F + ch]);
}

template <int F>
__global__ void bias_relu_kernel(float* __restrict__ y, const float* __restrict__ b,
                                 long long total) {
  long long i = blockIdx.x * (long long)blockDim.x + threadIdx.x;
  if (i >= total) return;
  int c = (int)i & (F - 1);
  float v = y[i] + b[c];
  y[i] = v > 0.f ? v : 0.f;
}

template <int F>
__global__ void bias_add_kernel(float* __restrict__ y, const float* __restrict__ b,
                                long long total) {
  long long i = blockIdx.x * (long long)blockDim.x + threadIdx.x;
  if (i >= total) return;
  int c = (int)i & (F - 1);
  y[i] = y[i] + b[c];
}

// ---------------------------------------------------------------------------
static inline long long cdiv(long long a, long long b) { return (a + b - 1) / b; }

extern "C" void kernel_launch(void* const* d_in, const int* in_sizes, int n_in,
                              void* d_out, int out_size, void* d_ws, size_t ws_size,
                              hipStream_t stream) {
  (void)n_in; (void)out_size; (void)ws_size;
  const float* x        = (const float*)d_in[0];
  const int*   ei       = (const int*)d_in[1];   // [2, E] int32
  const float* W1       = (const float*)d_in[2];
  const float* att_src1 = (const float*)d_in[3];
  const float* att_dst1 = (const float*)d_in[4];
  const float* b1       = (const float*)d_in[5];
  const float* W2       = (const float*)d_in[6];
  const float* att_src2 = (const float*)d_in[7];
  const float* att_dst2 = (const float*)d_in[8];
  const float* b2       = (const float*)d_in[9];

  constexpr int F_in = 128, H1 = 4, C1 = 32, HID = 128, F_out = 64;
  const int N = in_sizes[0] / F_in;
  const int E = in_sizes[1] / 2;
  const int Etot = E + N;                 // self-loops appended
  const int* srcI = ei;
  const int* dstI = ei + E;

  // workspace carve-up
  char* p = (char*)d_ws;
  auto alloc = [&](size_t bytes) -> char* {
    char* r = p;
    p += (bytes + 255) & ~(size_t)255;
    return r;
  };
  float*    h1    = (float*)alloc((size_t)N * HID * 4);   // layer-1 features; reused as h2
  float*    out1  = (float*)alloc((size_t)N * HID * 4);   // layer-1 aggregate -> layer-2 input
  float*    as1   = (float*)alloc((size_t)N * H1 * 4);    // reused for layer 2 (H=1)
  float*    ad1   = (float*)alloc((size_t)N * H1 * 4);
  unsigned* m1    = (unsigned*)alloc((size_t)N * H1 * 4);
  float*    den1  = (float*)alloc((size_t)N * H1 * 4);
  float*    alpha = (float*)alloc((size_t)Etot * H1 * 4);
  float*    outF  = (float*)d_out;                        // [N, 64]

  const int TB = 256;

  // ---------------- Layer 1 (H=4, C=32) ----------------
  hipMemsetAsync(m1,   0, (size_t)N * H1 * 4, stream);
  hipMemsetAsync(den1, 0, (size_t)N * H1 * 4, stream);
  hipMemsetAsync(out1, 0, (size_t)N * HID * 4, stream);

  gemm_wmma_f32<HID><<<(N + 15) / 16, HID * 2, 0, stream>>>(x, W1, h1, N);

  attn_scores_kernel<H1, C1><<<cdiv((long long)N * H1, TB), TB, 0, stream>>>(
      h1, att_src1, att_dst1, as1, ad1, N);

  edge_alpha_max_kernel<H1><<<cdiv((long long)Etot * H1, TB), TB, 0, stream>>>(
      srcI, dstI, as1, ad1, alpha, m1, E, Etot);
  edge_exp_sum_kernel<H1><<<cdiv((long long)Etot * H1, TB), TB, 0, stream>>>(
      dstI, m1, alpha, den1, E, Etot);
  edge_norm_kernel<H1><<<cdiv((long long)Etot * H1, TB), TB, 0, stream>>>(
      dstI, den1, alpha, E, Etot);
  edge_aggregate_kernel<H1, C1><<<cdiv((long long)Etot * HID, TB), TB, 0, stream>>>(
      srcI, dstI, alpha, h1, out1, E, Etot);

  bias_relu_kernel<HID><<<cdiv((long long)N * HID, TB), TB, 0, stream>>>(
      out1, b1, (long long)N * HID);

  // ---------------- Layer 2 (H=1, C=64) ----------------
  hipMemsetAsync(m1,   0, (size_t)N * 4, stream);
  hipMemsetAsync(den1, 0, (size_t)N * 4, stream);
  hipMemsetAsync(outF, 0, (size_t)N * F_out * 4, stream);

  float* h2 = h1;  // reuse: layer-1 h no longer needed
  gemm_wmma_f32<F_out><<<(N + 15) / 16, F_out * 2, 0, stream>>>(out1, W2, h2, N);

  attn_scores_kernel<1, F_out><<<cdiv((long long)N, TB), TB, 0, stream>>>(
      h2, att_src2, att_dst2, as1, ad1, N);

  edge_alpha_max_kernel<1><<<cdiv((long long)Etot, TB), TB, 0, stream>>>(
      srcI, dstI, as1, ad1, alpha, m1, E, Etot);
  edge_exp_sum_kernel<1><<<cdiv((long long)Etot, TB), TB, 0, stream>>>(
      dstI, m1, alpha, den1, E, Etot);
  edge_norm_kernel<1><<<cdiv((long long)Etot, TB), TB, 0, stream>>>(
      dstI, den1, alpha, E, Etot);
  edge_aggregate_kernel<1, F_out><<<cdiv((long long)Etot * F_out, TB), TB, 0, stream>>>(
      srcI, dstI, alpha, h2, outF, E, Etot);

  bias_add_kernel<F_out><<<cdiv((long long)N * F_out, TB), TB, 0, stream>>>(
      outF, b2, (long long)N * F_out);
}